// PointEncoder_31353261261653
// MI455X (gfx1250) — compile-verified
//
#include <hip/hip_runtime.h>

#define BN_EPS 1e-5f

typedef float v2f __attribute__((ext_vector_type(2)));
typedef float v8f __attribute__((ext_vector_type(8)));
typedef int   v4i __attribute__((ext_vector_type(4)));
typedef int   v8i __attribute__((ext_vector_type(8)));
typedef unsigned int v4u __attribute__((ext_vector_type(4)));

static __host__ __device__ inline int pad16(int x) { return (x + 15) & ~15; }

// ---------------------------------------------------------------------------
// Farthest point sampling: one block per batch. dist kept in LDS.
// Matches jax scan: record 'far' BEFORE the distance update; argmax = first max.
// ---------------------------------------------------------------------------
__global__ __launch_bounds__(256) void fps_kernel(const float* __restrict__ xyz,
                                                  int n, int npoint,
                                                  int* __restrict__ out) {
  const int b = blockIdx.x;
  const int t = threadIdx.x;
  const float* px = xyz + (size_t)b * 3 * n;
  const float* py = px + n;
  const float* pz = px + 2 * n;
  __shared__ float dist[2048];
  __shared__ float rv[256];
  __shared__ int   ri[256];
  __shared__ int   far_s;
  for (int i = t; i < n; i += 256) dist[i] = 1e10f;
  if (t == 0) far_s = 0;
  __syncthreads();
  for (int it = 0; it < npoint; ++it) {
    const int far = far_s;
    if (t == 0) out[b * npoint + it] = far;
    const float cx = px[far], cy = py[far], cz = pz[far];
    float best = -1.0f;
    int   bi   = 0;
    for (int i = t; i < n; i += 256) {
      float dx = px[i] - cx, dy = py[i] - cy, dz = pz[i] - cz;
      float d  = dx * dx + dy * dy + dz * dz;
      float nd = fminf(dist[i], d);
      dist[i]  = nd;
      if (nd > best) { best = nd; bi = i; }   // strict > keeps lowest index
    }
    rv[t] = best; ri[t] = bi;
    __syncthreads();
    for (int off = 128; off > 0; off >>= 1) {
      if (t < off) {
        if (rv[t + off] > rv[t] || (rv[t + off] == rv[t] && ri[t + off] < ri[t])) {
          rv[t] = rv[t + off]; ri[t] = ri[t + off];
        }
      }
      __syncthreads();
    }
    if (t == 0) far_s = ri[0];
    __syncthreads();
  }
}

// new_xyz (B,3,s) gathered from xyz (B,3,n) at fps indices
__global__ void gather_newxyz(const float* __restrict__ xyz, const int* __restrict__ fidx,
                              int n, int s, float* __restrict__ nxyz) {
  int gid = blockIdx.x * blockDim.x + threadIdx.x;
  if (gid >= 8 * 3 * s) return;
  int ss = gid % s;
  int c  = (gid / s) % 3;
  int b  = gid / (3 * s);
  int j  = fidx[b * s + ss];
  nxyz[gid] = xyz[((size_t)b * 3 + c) * n + j];
}

// Ball query: first k ascending indices within radius, pad with first hit.
__global__ void ballquery(const float* __restrict__ xyz, const float* __restrict__ nxyz,
                          int n, int s, int k, float r2, int* __restrict__ idx) {
  int gid = blockIdx.x * blockDim.x + threadIdx.x;
  if (gid >= 8 * s) return;
  int b  = gid / s;
  int ss = gid % s;
  const float* px = xyz + (size_t)b * 3 * n;
  const float* py = px + n;
  const float* pz = px + 2 * n;
  float cx = nxyz[((size_t)b * 3 + 0) * s + ss];
  float cy = nxyz[((size_t)b * 3 + 1) * s + ss];
  float cz = nxyz[((size_t)b * 3 + 2) * s + ss];
  int* o = idx + ((size_t)b * s + ss) * k;
  int cnt = 0, first = -1;
  for (int i = 0; i < n && cnt < k; ++i) {
    float dx = px[i] - cx, dy = py[i] - cy, dz = pz[i] - cz;
    if (dx * dx + dy * dy + dz * dz <= r2) {
      if (first < 0) first = i;
      o[cnt++] = i;
    }
  }
  if (first < 0) first = 0;
  for (; cnt < k; ++cnt) o[cnt] = first;
}

// Build grouped tensor X[in_pad][N], N = 8*k*s, column n = b*k*s + kk*s + ss.
__global__ void group_kernel(const float* __restrict__ pts, const float* __restrict__ xyz,
                             const float* __restrict__ nxyz, const int* __restrict__ idx,
                             float* __restrict__ X, int Cp, int n, int s, int k,
                             int in_pad, long long N) {
  long long gid = (long long)blockIdx.x * blockDim.x + threadIdx.x;
  if (gid >= N) return;
  int ss  = (int)(gid % s);
  int kk  = (int)((gid / s) % k);
  int b   = (int)(gid / ((long long)k * s));
  int j   = idx[((size_t)b * s + ss) * k + kk];
  for (int d = 0; d < Cp; ++d)
    X[(long long)d * N + gid] = pts[((size_t)b * Cp + d) * n + j];
  for (int c = 0; c < 3; ++c) {
    float v = xyz[((size_t)b * 3 + c) * n + j] - nxyz[((size_t)b * 3 + c) * s + ss];
    X[(long long)(Cp + c) * N + gid] = v;
  }
  for (int d = Cp + 3; d < in_pad; ++d)
    X[(long long)d * N + gid] = 0.0f;
}

// Zero-pad W (out_r x in_r) -> Wp (out_p x in_p); bias likewise.
__global__ void padW_kernel(const float* __restrict__ W, const float* __restrict__ bias,
                            float* __restrict__ Wp, float* __restrict__ bp,
                            int out_r, int in_r, int out_p, int in_p) {
  int gid = blockIdx.x * blockDim.x + threadIdx.x;
  if (gid >= out_p * in_p) return;
  int o = gid / in_p;
  int d = gid % in_p;
  Wp[gid] = (o < out_r && d < in_r) ? W[(size_t)o * in_r + d] : 0.0f;
  if (d == 0) bp[o] = (o < out_r) ? bias[o] : 0.0f;
}

// ---------------------------------------------------------------------------
// FP32 WMMA GEMM with double-buffered TDM-staged B operand.
//   Block = 128 threads (4 waves). One block owns a 16-column strip of X/Y
//   and ALL M tiles: wave w computes tiles {w, w+4, ...} (<= MT each).
//   X strip staged into LDS GEMM_KC K-rows at a time by the Tensor Data Mover
//   (2-D descriptor: tile 16 x chunk, row stride N elements). Two LDS buffers:
//   wave 0 issues the TDM for chunk c+1 before waiting for chunk c, so the DMA
//   overlaps the WMMA work on the current chunk. Per chunk, the M-tile loop is
//   OUTSIDE the branch-free K-loop so loads can pipeline across k iterations.
// A-frag (32-bit 16x4): lanes 0-15 -> M=lane, K {k0,k0+1}; lanes 16-31 ->
// M=lane-16, K {k0+2,k0+3}. B-frag mirrored over N. C/D: v8f per ISA layout.
// ---------------------------------------------------------------------------
#define GEMM_KC 128

__device__ __forceinline__ void tdm_stage(const float* __restrict__ X, long long n0,
                                          long long N, int Kpad, int kbase, int chunk,
                                          unsigned lds_off) {
  unsigned long long gaddr =
      (unsigned long long)(const void*)(X + (long long)kbase * N + n0);
  v4u g0;
  g0.x = 1u;                                    // count=1, user descriptor
  g0.y = lds_off;                               // LDS byte address of dest buffer
  g0.z = (unsigned)(gaddr & 0xFFFFFFFFu);       // global_addr[31:0]
  g0.w = (unsigned)((gaddr >> 32) & 0x01FFFFFFu) | (2u << 30);  // addr[56:32], type=2
  unsigned nd0 = (unsigned)N;                   // tensor_dim0 (elements)
  unsigned nd1 = (unsigned)Kpad;                // tensor_dim1
  unsigned long long st0 = (unsigned long long)N;  // dim0 stride (elements)
  v8i g1;
  g1[0] = (int)(2u << 16);                      // data_size=4B, mask=0, no flags
  g1[1] = (int)((nd0 & 0xFFFFu) << 16);         // dim0[15:0] -> bits 63:48
  g1[2] = (int)(((nd0 >> 16) & 0xFFFFu) | ((nd1 & 0xFFFFu) << 16));
  g1[3] = (int)(((nd1 >> 16) & 0xFFFFu) | (16u << 16));   // tile_dim0 = 16
  g1[4] = (int)((unsigned)chunk & 0xFFFFu);     // tile_dim1 = chunk, tile_dim2 = 0
  g1[5] = (int)(st0 & 0xFFFFFFFFu);             // dim0_stride[31:0]
  g1[6] = (int)((st0 >> 32) & 0xFFFFu);         // dim0_stride[47:32], dim1_stride lo = 0
  g1[7] = 0;
  v4i z4 = {0, 0, 0, 0};
#if __clang_major__ >= 23
  v8i z8 = {0, 0, 0, 0, 0, 0, 0, 0};
  __builtin_amdgcn_tensor_load_to_lds(g0, g1, z4, z4, z8, 0);
#else
  __builtin_amdgcn_tensor_load_to_lds(g0, g1, z4, z4, 0);
#endif
}

template <int MT>
__global__ __launch_bounds__(128) void gemm_wmma_t(const float* __restrict__ A,
                                                   const float* __restrict__ X,
                                                   const float* __restrict__ bias,
                                                   float* __restrict__ Y,
                                                   int Kpad, int Mtiles, long long N) {
  __shared__ float bs[2][GEMM_KC * 16];   // sole LDS object -> base offset 0
  const int t    = threadIdx.x;
  const int lane = t & 31;
  // Scalarize the wave id: all tile guards become SGPR compares (no EXEC
  // masking), so EXEC stays all-ones around every WMMA.
  const int wq   = __builtin_amdgcn_readfirstlane(t >> 5);
  const int lm   = lane & 15;
  const int half = lane >> 4;
  const long long n0 = (long long)blockIdx.x * 16;
  const int nch  = (Kpad + GEMM_KC - 1) / GEMM_KC;
  // number of M tiles this wave owns (scalar, hoisted out of all loops)
  const int icnt = (Mtiles - wq + 3) >> 2;

  v8f acc[MT];
#pragma unroll
  for (int i = 0; i < MT; ++i) acc[i] = (v8f){};

  if (wq == 0) {                    // prime the pipeline: stage chunk 0
    const int chunk0 = (Kpad < GEMM_KC) ? Kpad : GEMM_KC;
    tdm_stage(X, n0, N, Kpad, 0, chunk0, 0u);
  }

  for (int c = 0; c < nch; ++c) {
    const int kbase = c * GEMM_KC;
    const int chunk = (Kpad - kbase < GEMM_KC) ? (Kpad - kbase) : GEMM_KC;

    if (wq == 0) {
      if (c + 1 < nch) {
        // Buffer (c+1)&1 was last read in iteration c-1; the end-of-iteration
        // barrier below guarantees those reads are done, so prefetch now.
        const int kb1 = kbase + GEMM_KC;
        const int ch1 = (Kpad - kb1 < GEMM_KC) ? (Kpad - kb1) : GEMM_KC;
        tdm_stage(X, n0, N, Kpad, kb1, ch1, (unsigned)(((c + 1) & 1) * (GEMM_KC * 64)));
        __builtin_amdgcn_s_wait_tensorcnt(1);   // chunk c resident (in-order)
      } else {
        __builtin_amdgcn_s_wait_tensorcnt(0);
      }
    }
    __syncthreads();                 // release consumers for chunk c

    const float* bsc = bs[c & 1];
#pragma unroll
    for (int i = 0; i < MT; ++i) {
      if (i < icnt) {                // scalar, loop-invariant in k
        const float* ar =
            A + (size_t)((wq + 4 * i) * 16 + lm) * Kpad + kbase + 2 * half;
        v8f a = acc[i];
#pragma unroll 4
        for (int k0 = 0; k0 < chunk; k0 += 4) {   // branch-free inner loop
          v2f bv;
          bv.x = bsc[(k0 + 2 * half) * 16 + lm];
          bv.y = bsc[(k0 + 2 * half + 1) * 16 + lm];
          v2f av;
          av.x = ar[k0];
          av.y = ar[k0 + 1];
          a = __builtin_amdgcn_wmma_f32_16x16x4_f32(false, av, false, bv,
                                                    (short)0, a, false, false);
        }
        acc[i] = a;
      }
    }
    __syncthreads();                 // everyone done reading bs[c&1]
  }

  const long long col = n0 + lm;
#pragma unroll
  for (int i = 0; i < MT; ++i) {
    if (i < icnt) {
      const int mt = wq + 4 * i;
      for (int r = 0; r < 8; ++r) {
        int m = mt * 16 + r + 8 * half;
        Y[(long long)m * N + col] = acc[i][r] + bias[m];
      }
    }
  }
}

// Per-channel mean/var over N elements (population variance, matches jnp.var).
__global__ __launch_bounds__(256) void bn_stats(const float* __restrict__ Y, long long N,
                                                float* __restrict__ mean,
                                                float* __restrict__ var) {
  const int o = blockIdx.x;
  const int t = threadIdx.x;
  __shared__ float ssum[256];
  __shared__ float ssq[256];
  const float4* y4 = (const float4*)(Y + (long long)o * N);
  const long long n4 = N >> 2;
  float a = 0.0f, q = 0.0f;
  for (long long i = t; i < n4; i += 256) {
    float4 v = y4[i];
    a += v.x + v.y + v.z + v.w;
    q += v.x * v.x + v.y * v.y + v.z * v.z + v.w * v.w;
  }
  ssum[t] = a; ssq[t] = q;
  __syncthreads();
  for (int off = 128; off > 0; off >>= 1) {
    if (t < off) { ssum[t] += ssum[t + off]; ssq[t] += ssq[t + off]; }
    __syncthreads();
  }
  if (t == 0) {
    float m = ssum[0] / (float)N;
    mean[o] = m;
    var[o]  = ssq[0] / (float)N - m * m;
  }
}

__global__ void bn_relu(float* __restrict__ Y, const float* __restrict__ mean,
                        const float* __restrict__ var, const float* __restrict__ gamma,
                        const float* __restrict__ beta, int oc, long long N) {
  const long long n4 = N >> 2;
  long long gid = (long long)blockIdx.x * blockDim.x + threadIdx.x;
  if (gid >= (long long)oc * n4) return;
  int o = (int)(gid / n4);
  float sc = rsqrtf(var[o] + BN_EPS) * gamma[o];
  float mu = mean[o], be = beta[o];
  float4* y4 = (float4*)Y;
  float4 v = y4[gid];
  v.x = fmaxf((v.x - mu) * sc + be, 0.0f);
  v.y = fmaxf((v.y - mu) * sc + be, 0.0f);
  v.z = fmaxf((v.z - mu) * sc + be, 0.0f);
  v.w = fmaxf((v.w - mu) * sc + be, 0.0f);
  y4[gid] = v;
}

// max over k samples; write into concatenated l_points slot (B, Ctot, s)
__global__ void maxpool_kernel(const float* __restrict__ Y, float* __restrict__ outp,
                               int oc, int k, int s, int c_off, int Ctot, long long N) {
  long long gid = (long long)blockIdx.x * blockDim.x + threadIdx.x;
  if (gid >= 8LL * oc * s) return;
  int ss = (int)(gid % s);
  int o  = (int)((gid / s) % oc);
  int b  = (int)(gid / ((long long)oc * s));
  float m = -1e30f;
  const float* y = Y + (long long)o * N + (long long)b * k * s + ss;
  for (int kk = 0; kk < k; ++kk) m = fmaxf(m, y[(long long)kk * s]);
  outp[((size_t)b * Ctot + c_off + o) * s + ss] = m;
}

// ---------------------------------------------------------------------------
// Host orchestration
// ---------------------------------------------------------------------------
static inline size_t align_up(size_t x) { return (x + 255) & ~(size_t)255; }

static void launch_gemm(dim3 grid, hipStream_t stream, const float* Wp, const float* X,
                        const float* bp, float* Y, int Kpad, int Mtiles, long long N) {
  const int MTmax = (Mtiles + 3) / 4;
  switch (MTmax) {
    case 1: gemm_wmma_t<1><<<grid, 128, 0, stream>>>(Wp, X, bp, Y, Kpad, Mtiles, N); break;
    case 2: gemm_wmma_t<2><<<grid, 128, 0, stream>>>(Wp, X, bp, Y, Kpad, Mtiles, N); break;
    case 3: gemm_wmma_t<3><<<grid, 128, 0, stream>>>(Wp, X, bp, Y, Kpad, Mtiles, N); break;
    default: gemm_wmma_t<4><<<grid, 128, 0, stream>>>(Wp, X, bp, Y, Kpad, Mtiles, N); break;
  }
}

extern "C" void kernel_launch(void* const* d_in, const int* in_sizes, int n_in,
                              void* d_out, int out_size, void* d_ws, size_t ws_size,
                              hipStream_t stream) {
  (void)in_sizes; (void)n_in; (void)out_size; (void)ws_size;
  const float* xyz_in = (const float*)d_in[0];
  float* out = (float*)d_out;
  char*  ws  = (char*)d_ws;

  // output offsets (floats)
  const size_t L0X = 0,        L0P = 49152;
  const size_t L1X = 98304,    L1P = 110592;
  const size_t L2X = 1421312,  L2P = 1424384;
  const size_t L3X = 1948672,  L3P = 1950208;

  // workspace layout
  size_t off = 0;
  size_t off_fps = off;                 off = align_up(off + (size_t)8 * 512 * 4);
  size_t off_idx = off;                 off = align_up(off + (size_t)8 * 512 * 128 * 4);
  size_t off_wp  = off;                 off = align_up(off + (size_t)256 * 528 * 4);
  size_t off_bp  = off;                 off = align_up(off + (size_t)256 * 4);
  size_t off_mu  = off;                 off = align_up(off + (size_t)256 * 4);
  size_t off_var = off;                 off = align_up(off + (size_t)256 * 4);
  size_t off_bufA = off;                off = align_up(off + (size_t)50331648 * 4); // 96*524288
  size_t off_bufB = off;                off = align_up(off + (size_t)67108864 * 4); // 128*524288

  int*   fps_idx = (int*)(ws + off_fps);
  int*   bq_idx  = (int*)(ws + off_idx);
  float* Wp      = (float*)(ws + off_wp);
  float* bp      = (float*)(ws + off_bp);
  float* mu      = (float*)(ws + off_mu);
  float* var     = (float*)(ws + off_var);
  float* bufA    = (float*)(ws + off_bufA);
  float* bufB    = (float*)(ws + off_bufB);

  // l0 outputs: identity copies of input xyz
  hipMemcpyAsync(out + L0X, xyz_in, (size_t)49152 * 4, hipMemcpyDeviceToDevice, stream);
  hipMemcpyAsync(out + L0P, xyz_in, (size_t)49152 * 4, hipMemcpyDeviceToDevice, stream);

  // level tables
  const int   n_arr[3]    = {2048, 512, 128};
  const int   s_arr[3]    = {512, 128, 64};
  const int   cpts_arr[3] = {3, 320, 512};
  const int   ctot_arr[3] = {320, 512, 512};
  const int   nbr_arr[3]  = {3, 2, 2};
  const float rad[3][3]   = {{0.1f, 0.2f, 0.4f}, {0.4f, 0.8f, 0.0f}, {0.2f, 0.4f, 0.0f}};
  const int   ks[3][3]    = {{32, 64, 128}, {64, 128, 0}, {16, 32, 0}};
  const int   mlp[3][3][3] = {{{32, 32, 64}, {64, 64, 128}, {64, 96, 128}},
                              {{128, 128, 256}, {128, 196, 256}, {0, 0, 0}},
                              {{128, 128, 256}, {128, 196, 256}, {0, 0, 0}}};
  const float* lvl_xyz[3] = {xyz_in, out + L1X, out + L2X};
  const float* lvl_pts[3] = {xyz_in, out + L1P, out + L2P};
  float* out_xyz[3] = {out + L1X, out + L2X, out + L3X};
  float* out_pts[3] = {out + L1P, out + L2P, out + L3P};

  int layer_counter = 0;
  for (int L = 0; L < 3; ++L) {
    const int n = n_arr[L], s = s_arr[L], Cp = cpts_arr[L], Ctot = ctot_arr[L];

    fps_kernel<<<8, 256, 0, stream>>>(lvl_xyz[L], n, s, fps_idx);
    gather_newxyz<<<(8 * 3 * s + 255) / 256, 256, 0, stream>>>(lvl_xyz[L], fps_idx, n, s,
                                                               out_xyz[L]);
    int c_off = 0;
    for (int br = 0; br < nbr_arr[L]; ++br) {
      const int k = ks[L][br];
      const long long N = 8LL * k * s;
      const float r2 = rad[L][br] * rad[L][br];

      ballquery<<<(8 * s + 255) / 256, 256, 0, stream>>>(lvl_xyz[L], out_xyz[L], n, s, k,
                                                         r2, bq_idx);
      const int in_real0 = Cp + 3;
      const int in_pad0  = pad16(in_real0);
      group_kernel<<<(int)((N + 255) / 256), 256, 0, stream>>>(lvl_pts[L], lvl_xyz[L],
                                                               out_xyz[L], bq_idx, bufA,
                                                               Cp, n, s, k, in_pad0, N);
      float* cur = bufA;
      float* nxt = bufB;
      int prev_real = in_real0, prev_pad = in_pad0;
      for (int l = 0; l < 3; ++l) {
        const int oc     = mlp[L][br][l];
        const int oc_pad = pad16(oc);
        const float* W     = (const float*)d_in[1 + 4 * layer_counter + 0];
        const float* bias  = (const float*)d_in[1 + 4 * layer_counter + 1];
        const float* gamma = (const float*)d_in[1 + 4 * layer_counter + 2];
        const float* beta  = (const float*)d_in[1 + 4 * layer_counter + 3];
        ++layer_counter;

        padW_kernel<<<(oc_pad * prev_pad + 255) / 256, 256, 0, stream>>>(
            W, bias, Wp, bp, oc, prev_real, oc_pad, prev_pad);

        dim3 g((unsigned)(N / 16));
        launch_gemm(g, stream, Wp, cur, bp, nxt, prev_pad, oc_pad / 16, N);

        bn_stats<<<oc, 256, 0, stream>>>(nxt, N, mu, var);
        bn_relu<<<(int)((((long long)oc * N) / 4 + 255) / 256), 256, 0, stream>>>(
            nxt, mu, var, gamma, beta, oc, N);

        float* t = cur; cur = nxt; nxt = t;
        prev_real = oc; prev_pad = oc_pad;
      }
      maxpool_kernel<<<(int)((8LL * prev_real * s + 255) / 256), 256, 0, stream>>>(
          cur, out_pts[L], prev_real, k, s, c_off, Ctot, N);
      c_off += prev_real;
    }
  }
}